// SparseSAKEEnergyModel_20538533609906
// MI455X (gfx1250) — compile-verified
//
#include <hip/hip_runtime.h>

// ---------------------------------------------------------------------------
// SAKE GNN forward for MI455X (gfx1250, wave32).
// GEMMs on the fp32 matrix pipe (V_WMMA_F32_16X16X4_F32), with CDNA5
// async global->LDS staging (GLOBAL_LOAD_ASYNC_TO_LDS_B128/B32, ASYNCcnt)
// and double-buffered K-chunks to hide HBM latency behind the WMMA pipe.
// Segment softmax / segment sums via L2 atomics.
// ---------------------------------------------------------------------------

typedef float v2f __attribute__((ext_vector_type(2)));
typedef float v8f __attribute__((ext_vector_type(8)));

#define NN      10000       // nodes
#define NE      100000      // edges
#define FIN     16
#define HDIM    64
#define NHEADS  4
#define NCOEF   256         // HDIM*NHEADS
#define NDEPTH  6
#define NRBF    50
#define NSEGS   100
#define HEIN    179         // 2H + RBF + 1
#define HEINP   192         // padded row stride
#define NODEIN  384         // H + COEF + H

#define ASTR    36          // LDS A stride: 16B-aligned rows, gcd(36,64)=4 -> 16 distinct banks
#define BSTR    80          // LDS B stride: bank=(16*row+n)%64 -> half-waves hit disjoint banks

static __device__ __forceinline__ float siluf(float x) { return x / (1.f + expf(-x)); }
// monotone float <-> uint encoding for atomicMax on floats
static __device__ __forceinline__ unsigned fenc(float f) {
  unsigned u = __float_as_uint(f);
  return (u & 0x80000000u) ? ~u : (u | 0x80000000u);
}
static __device__ __forceinline__ float fdec(unsigned u) {
  unsigned b = (u & 0x80000000u) ? (u & 0x7FFFFFFFu) : ~u;
  return __uint_as_float(b);
}

// CDNA5 async global->LDS copies (tracked by ASYNCcnt).
// LDS operand = low 32 bits of the generic pointer (shared aperture encoding).
static __device__ __forceinline__ void async_b128(float* lds, const float* g) {
  unsigned l = (unsigned)(uintptr_t)(void*)lds;
  unsigned long long a = (unsigned long long)(uintptr_t)(const void*)g;
  asm volatile("global_load_async_to_lds_b128 %0, %1, off" :: "v"(l), "v"(a) : "memory");
}
static __device__ __forceinline__ void async_b32(float* lds, const float* g) {
  unsigned l = (unsigned)(uintptr_t)(void*)lds;
  unsigned long long a = (unsigned long long)(uintptr_t)(const void*)g;
  asm volatile("global_load_async_to_lds_b32 %0, %1, off" :: "v"(l), "v"(a) : "memory");
}
static __device__ __forceinline__ void wait_async0() {
  asm volatile("s_wait_asynccnt 0x0" ::: "memory");
}

// ---------------------------------------------------------------------------
// Generic fp32 WMMA GEMM: C[M,N] = act(A[M,K] @ B[K,N] + bias) (+ resid)
// Block = 256 threads = 8 wave32 waves; block tile 64x64; K chunked by 32,
// double-buffered via async LDS loads.  act: 0=none, 1=silu, 2=tanh
// ---------------------------------------------------------------------------
__global__ __launch_bounds__(256)
void gemm_wmma_f32(const float* __restrict__ A, int lda,
                   const float* __restrict__ B, int ldb,
                   const float* __restrict__ bias,
                   const float* __restrict__ resid,
                   float* __restrict__ C, int ldc,
                   int M, int N, int K, int act)
{
  __shared__ float As[2][64 * ASTR];
  __shared__ float Bs[2][32 * BSTR];

  const int tid  = threadIdx.x;
  const int lane = tid & 31;
  const int wave = tid >> 5;
  const int mi   = wave & 3;       // m subtile 0..3
  const int nio  = wave >> 2;      // 0..1 -> n subtiles {2*nio, 2*nio+1}
  const int hv   = lane >> 4;      // half-wave: 0 or 1
  const int l16  = lane & 15;

  const int bm = blockIdx.y * 64;
  const int bn = blockIdx.x * 64;

  v8f acc0 = {};
  v8f acc1 = {};

  auto load_chunk = [&](int buf, int kc) {
    // A tile 64x32: 512 16-byte chunks, 2 per thread (lda is always 16B-mult)
    #pragma unroll
    for (int i = 0; i < 2; ++i) {
      int idx = tid + i * 256;
      int r  = idx >> 3;
      int c4 = (idx & 7) * 4;
      int gr = bm + r, gk = kc + c4;
      float* dst = &As[buf][r * ASTR + c4];
      if (gr < M && gk + 3 < K) {
        async_b128(dst, &A[(size_t)gr * lda + gk]);
      } else {
        #pragma unroll
        for (int q = 0; q < 4; ++q)
          dst[q] = (gr < M && gk + q < K) ? A[(size_t)gr * lda + gk + q] : 0.f;
      }
    }
    // B tile 32x64: per-element b32 asyncs (ldb may be odd, e.g. 50)
    #pragma unroll
    for (int i = 0; i < 8; ++i) {
      int idx = tid + i * 256;
      int r = idx >> 6, c = idx & 63;
      int gk = kc + r, gc = bn + c;
      float* dst = &Bs[buf][r * BSTR + c];
      if (gk < K && gc < N) async_b32(dst, &B[(size_t)gk * ldb + gc]);
      else                  *dst = 0.f;
    }
  };

  const int nchunks = (K + 31) >> 5;
  load_chunk(0, 0);
  wait_async0();
  __syncthreads();

  const int n0 = (nio * 2) * 16;
  const int n1 = n0 + 16;

  for (int ci = 0; ci < nchunks; ++ci) {
    const int buf = ci & 1;
    if (ci + 1 < nchunks) load_chunk(buf ^ 1, (ci + 1) << 5);  // prefetch next

    #pragma unroll
    for (int k = 0; k < 32; k += 4) {
      // f32 16x16x4 fragment layout:
      // A: VGPR0 = K=k (lanes 0-15) / K=k+2 (lanes 16-31); VGPR1 = k+1 / k+3
      // B: same K split; N = lane&15.  C/D: row = r + 8*half, col = lane&15.
      int kb = k + hv * 2;
      v2f a, b0, b1;
      a.x  = As[buf][(mi * 16 + l16) * ASTR + kb];
      a.y  = As[buf][(mi * 16 + l16) * ASTR + kb + 1];
      b0.x = Bs[buf][kb * BSTR + n0 + l16];
      b0.y = Bs[buf][(kb + 1) * BSTR + n0 + l16];
      b1.x = Bs[buf][kb * BSTR + n1 + l16];
      b1.y = Bs[buf][(kb + 1) * BSTR + n1 + l16];
      acc0 = __builtin_amdgcn_wmma_f32_16x16x4_f32(false, a, false, b0, (short)0, acc0, false, false);
      acc1 = __builtin_amdgcn_wmma_f32_16x16x4_f32(false, a, false, b1, (short)0, acc1, false, false);
    }
    wait_async0();      // next chunk's async writes complete (overlapped with WMMAs)
    __syncthreads();
  }

  const int c0 = bn + n0 + l16;
  const int c1 = c0 + 16;
  #pragma unroll
  for (int r = 0; r < 8; ++r) {
    int row = bm + mi * 16 + r + hv * 8;
    if (row >= M) continue;
    if (c0 < N) {
      float v = acc0[r];
      if (bias) v += bias[c0];
      if (act == 1) v = siluf(v); else if (act == 2) v = tanhf(v);
      if (resid) v += resid[(size_t)row * ldc + c0];
      C[(size_t)row * ldc + c0] = v;
    }
    if (c1 < N) {
      float v = acc1[r];
      if (bias) v += bias[c1];
      if (act == 1) v = siluf(v); else if (act == 2) v = tanhf(v);
      if (resid) v += resid[(size_t)row * ldc + c1];
      C[(size_t)row * ldc + c1] = v;
    }
  }
}

// ---------------------------------------------------------------------------
// Elementwise / scatter kernels
// ---------------------------------------------------------------------------
__global__ void k_fill0(float* p, int n) {
  int i = blockIdx.x * blockDim.x + threadIdx.x;
  if (i < n) p[i] = 0.f;
}

__global__ void k_edge_geom(const int* __restrict__ edges, const float* __restrict__ x,
                            float* __restrict__ d, float* __restrict__ dirs,
                            float* __restrict__ rbfb) {
  int e = blockIdx.x * blockDim.x + threadIdx.x;
  if (e >= NE) return;
  int dst = edges[2 * e], src = edges[2 * e + 1];
  float dx = x[dst * 3 + 0] - x[src * 3 + 0];
  float dy = x[dst * 3 + 1] - x[src * 3 + 1];
  float dz = x[dst * 3 + 2] - x[src * 3 + 2];
  float dd = sqrtf(dx * dx + dy * dy + dz * dz + 1e-10f);
  float inv = 1.f / (dd + 1e-5f);
  d[e] = dd;
  dirs[e * 3 + 0] = dx * inv;
  dirs[e * 3 + 1] = dy * inv;
  dirs[e * 3 + 2] = dz * inv;
  rbfb[e] = expf(-dd);
}

// he_in[:, 0:128] = h[dst] || h[src]
__global__ void k_gather(const int* __restrict__ edges, const float* __restrict__ h,
                         float* __restrict__ hein) {
  int idx = blockIdx.x * blockDim.x + threadIdx.x;
  if (idx >= NE * 128) return;
  int e = idx >> 7, c = idx & 127;
  int node = (c < HDIM) ? edges[2 * e] : edges[2 * e + 1];
  hein[e * HEINP + c] = h[node * HDIM + (c & 63)];
}

// he_in[:,128:178]=rbf*hm, [:,178]=d, [:,179:192]=0
__global__ void k_rbf(const float* __restrict__ hm, const float* __restrict__ rbfb,
                      const float* __restrict__ d, const float* __restrict__ means,
                      const float* __restrict__ betas, float* __restrict__ hein) {
  int idx = blockIdx.x * blockDim.x + threadIdx.x;
  if (idx >= NE * 64) return;
  int e = idx >> 6, j = idx & 63;
  float v;
  if (j < NRBF) {
    float t = rbfb[e] - means[j];
    v = expf(-betas[j] * t * t) * hm[e * 64 + j];
  } else if (j == NRBF) {
    v = d[e];
  } else {
    v = 0.f;
  }
  hein[e * HEINP + 128 + j] = v;
}

// logits = celu(he @ sem_w + sem_b, alpha=2)
__global__ void k_sem(const float* __restrict__ he, const float* __restrict__ semw,
                      const float* __restrict__ semb, float* __restrict__ logits) {
  int idx = blockIdx.x * blockDim.x + threadIdx.x;
  if (idx >= NE * NHEADS) return;
  int e = idx >> 2, hd = idx & 3;
  float s = semb[hd];
  for (int j = 0; j < HDIM; ++j) s += he[e * HDIM + j] * semw[j * NHEADS + hd];
  logits[idx] = (s > 0.f) ? s : 2.f * (expf(s * 0.5f) - 1.f);
}

__global__ void k_att_max(const int* __restrict__ edges, const float* __restrict__ logits,
                          const float* __restrict__ d, unsigned* __restrict__ nmax_s,
                          unsigned* __restrict__ nmax_e) {
  int idx = blockIdx.x * blockDim.x + threadIdx.x;
  if (idx >= NE * NHEADS) return;
  int e = idx >> 2, hd = idx & 3;
  int dst = edges[2 * e];
  atomicMax(&nmax_s[dst * NHEADS + hd], fenc(logits[idx]));
  if (hd == 0) atomicMax(&nmax_e[dst], fenc(-d[e]));
}

__global__ void k_att_exp(const int* __restrict__ edges, const float* __restrict__ logits,
                          const float* __restrict__ d, const unsigned* __restrict__ nmax_s,
                          const unsigned* __restrict__ nmax_e, float* __restrict__ es,
                          float* __restrict__ ee, float* __restrict__ nsum_s,
                          float* __restrict__ nsum_e) {
  int idx = blockIdx.x * blockDim.x + threadIdx.x;
  if (idx >= NE * NHEADS) return;
  int e = idx >> 2, hd = idx & 3;
  int dst = edges[2 * e];
  float v = expf(logits[idx] - fdec(nmax_s[dst * NHEADS + hd]));
  es[idx] = v;
  atomicAdd(&nsum_s[dst * NHEADS + hd], v);
  if (hd == 0) {
    float w = expf(-d[e] - fdec(nmax_e[dst]));
    ee[e] = w;
    atomicAdd(&nsum_e[dst], w);
  }
}

__global__ void k_att_un(const int* __restrict__ edges, const float* __restrict__ es,
                         const float* __restrict__ ee, const float* __restrict__ nsum_s,
                         const float* __restrict__ nsum_e, float* __restrict__ att,
                         float* __restrict__ natt) {
  int idx = blockIdx.x * blockDim.x + threadIdx.x;
  if (idx >= NE * NHEADS) return;
  int e = idx >> 2, hd = idx & 3;
  int dst = edges[2 * e];
  float a = (ee[e] / nsum_e[dst]) * (es[idx] / nsum_s[dst * NHEADS + hd]);
  att[idx] = a;
  atomicAdd(&natt[dst * NHEADS + hd], a);
}

// hea[e, j*4+hd] = he[e,j] * att_norm[e,hd]
__global__ void k_hea(const int* __restrict__ edges, const float* __restrict__ he,
                      const float* __restrict__ att, const float* __restrict__ natt,
                      float* __restrict__ hea) {
  int idx = blockIdx.x * blockDim.x + threadIdx.x;
  if (idx >= NE * NCOEF) return;
  int e = idx >> 8, c = idx & 255;
  int j = c >> 2, hd = c & 3;
  int dst = edges[2 * e];
  hea[idx] = he[e * HDIM + j] * (att[e * NHEADS + hd] / natt[dst * NHEADS + hd]);
}

// h_agg += hea ; csum[dst,c,dim] += coef*dirs
__global__ void k_scatter(const int* __restrict__ edges, const float* __restrict__ hea,
                          const float* __restrict__ coef, const float* __restrict__ dirs,
                          float* __restrict__ hagg, float* __restrict__ csum) {
  int idx = blockIdx.x * blockDim.x + threadIdx.x;
  if (idx >= NE * NCOEF) return;
  int e = idx >> 8, c = idx & 255;
  int dst = edges[2 * e];
  atomicAdd(&hagg[dst * NCOEF + c], hea[idx]);
  float cf = coef[idx];
  int base = (dst * NCOEF + c) * 3;
  atomicAdd(&csum[base + 0], cf * dirs[e * 3 + 0]);
  atomicAdd(&csum[base + 1], cf * dirs[e * 3 + 1]);
  atomicAdd(&csum[base + 2], cf * dirs[e * 3 + 2]);
}

__global__ void k_cnorm(const float* __restrict__ csum, float* __restrict__ cnorm) {
  int idx = blockIdx.x * blockDim.x + threadIdx.x;
  if (idx >= NN * NCOEF) return;
  float a = csum[idx * 3 + 0], b = csum[idx * 3 + 1], c = csum[idx * 3 + 2];
  cnorm[idx] = a * a + b * b + c * c;
}

__global__ void k_updin(const float* __restrict__ h, const float* __restrict__ hagg,
                        const float* __restrict__ hcomb, float* __restrict__ updin) {
  int idx = blockIdx.x * blockDim.x + threadIdx.x;
  if (idx >= NN * NODEIN) return;
  int n = idx / NODEIN, c = idx - n * NODEIN;
  float v;
  if (c < HDIM)              v = h[n * HDIM + c];
  else if (c < HDIM + NCOEF) v = hagg[n * NCOEF + (c - HDIM)];
  else                       v = hcomb[n * HDIM + (c - HDIM - NCOEF)];
  updin[idx] = v;
}

__global__ void k_yscatter(const int* __restrict__ segs, const float* __restrict__ hfin,
                           float* __restrict__ y) {
  int idx = blockIdx.x * blockDim.x + threadIdx.x;
  if (idx >= NN * HDIM) return;
  int n = idx >> 6, c = idx & 63;
  atomicAdd(&y[segs[n] * HDIM + c], hfin[idx]);
}

__global__ void k_head(const float* __restrict__ y, const float* __restrict__ w1,
                       const float* __restrict__ b1, const float* __restrict__ w2,
                       const float* __restrict__ b2, float* __restrict__ out) {
  int s = blockIdx.x * blockDim.x + threadIdx.x;
  if (s >= NSEGS) return;
  float acc = b2[0];
  for (int j = 0; j < HDIM; ++j) {
    float t = b1[j];
    for (int k = 0; k < HDIM; ++k) t += y[s * HDIM + k] * w1[k * HDIM + j];
    acc += siluf(t) * w2[j];
  }
  out[s] = acc;
}

// ---------------------------------------------------------------------------
// Host orchestration
// ---------------------------------------------------------------------------
extern "C" void kernel_launch(void* const* d_in, const int* in_sizes, int n_in,
                              void* d_out, int out_size, void* d_ws, size_t ws_size,
                              hipStream_t stream) {
  const float* in_i       = (const float*)d_in[0];
  const float* in_x       = (const float*)d_in[1];
  const float* emb_in_w   = (const float*)d_in[2];
  const float* emb_in_b   = (const float*)d_in[3];
  const float* edge_in_w  = (const float*)d_in[4];
  const float* edge_in_b  = (const float*)d_in[5];
  const float* rbf_means  = (const float*)d_in[6];
  const float* rbf_betas  = (const float*)d_in[7];
  const float* eo_w1      = (const float*)d_in[8];
  const float* eo_b1      = (const float*)d_in[9];
  const float* eo_w2      = (const float*)d_in[10];
  const float* eo_b2      = (const float*)d_in[11];
  const float* sem_w      = (const float*)d_in[12];
  const float* sem_b      = (const float*)d_in[13];
  const float* xmix_w     = (const float*)d_in[14];
  const float* pn_w1      = (const float*)d_in[15];
  const float* pn_b1      = (const float*)d_in[16];
  const float* pn_w2      = (const float*)d_in[17];
  const float* pn_b2      = (const float*)d_in[18];
  const float* node_w1    = (const float*)d_in[19];
  const float* node_b1    = (const float*)d_in[20];
  const float* node_w2    = (const float*)d_in[21];
  const float* node_b2    = (const float*)d_in[22];
  const float* emo_w1     = (const float*)d_in[23];
  const float* emo_b1     = (const float*)d_in[24];
  const float* emo_w2     = (const float*)d_in[25];
  const float* emo_b2     = (const float*)d_in[26];
  const float* head_w1    = (const float*)d_in[27];
  const float* head_b1    = (const float*)d_in[28];
  const float* head_w2    = (const float*)d_in[29];
  const float* head_b2    = (const float*)d_in[30];
  const int*   edges      = (const int*)d_in[31];
  const int*   segs       = (const int*)d_in[32];

  float* ws = (float*)d_ws;
  size_t off = 0;
  auto alloc = [&](size_t n) -> float* {
    float* p = ws + off;
    off += (n + 63) & ~(size_t)63;
    return p;
  };
  float*    hein   = alloc((size_t)NE * HEINP);   // 19.2M  (also aliased: coef, updin)
  float*    hm     = alloc((size_t)NE * 64);      // 6.4M   (hm then t1; tail of coef alias)
  float*    he     = alloc((size_t)NE * HDIM);    // 6.4M   (also aliased: cnorm)
  float*    hea    = alloc((size_t)NE * NCOEF);   // 25.6M
  float*    dbuf   = alloc(NE);
  float*    dirs   = alloc((size_t)NE * 3);
  float*    rbfb   = alloc(NE);
  float*    logits = alloc((size_t)NE * NHEADS);
  float*    es     = alloc((size_t)NE * NHEADS);
  float*    ee     = alloc(NE);
  float*    att    = alloc((size_t)NE * NHEADS);
  unsigned* nmax_s = (unsigned*)alloc((size_t)NN * NHEADS);   // start of att-accum span
  float*    nsum_s = alloc((size_t)NN * NHEADS);
  unsigned* nmax_e = (unsigned*)alloc(NN);
  float*    nsum_e = alloc(NN);
  float*    natt   = alloc((size_t)NN * NHEADS);              // end of att-accum span
  float*    csum   = alloc((size_t)NN * NCOEF * 3);
  float*    hagg   = alloc((size_t)NN * NCOEF);               // contiguous after csum
  float*    hbuf   = alloc((size_t)NN * HDIM);
  float*    tnode  = alloc((size_t)NN * HDIM);
  float*    hcomb  = alloc((size_t)NN * HDIM);
  float*    hfin   = alloc((size_t)NN * HDIM);
  float*    ybuf   = alloc((size_t)NSEGS * HDIM);
  float*    coefb  = hein;   // alive only between xmix GEMM and k_scatter
  float*    updin  = hein;   // alive only between k_updin and node GEMMs
  float*    cnorm  = he;     // he is dead once hea is built

  const int att_cnt  = (int)((natt + (size_t)NN * NHEADS) - (float*)nmax_s);
  const int node_cnt = (int)((hagg + (size_t)NN * NCOEF) - csum);

  auto gemm = [&](const float* A, int lda, const float* B, int ldb,
                  const float* bias, const float* resid, float* C, int ldc,
                  int M, int N, int K, int act) {
    dim3 g((unsigned)((N + 63) / 64), (unsigned)((M + 63) / 64));
    gemm_wmma_f32<<<g, 256, 0, stream>>>(A, lda, B, ldb, bias, resid, C, ldc, M, N, K, act);
  };
  auto blk = [](int n) { return dim3((unsigned)((n + 255) / 256)); };

  // --- setup ---
  k_edge_geom<<<blk(NE), 256, 0, stream>>>(edges, in_x, dbuf, dirs, rbfb);
  // h = i @ emb_in_w + b
  gemm(in_i, FIN, emb_in_w, HDIM, emb_in_b, nullptr, hbuf, HDIM, NN, HDIM, FIN, 0);

  for (int l = 0; l < NDEPTH; ++l) {
    const float* ei_w  = edge_in_w + (size_t)l * 128 * NRBF;
    const float* ei_b  = edge_in_b + (size_t)l * NRBF;
    const float* rm    = rbf_means + (size_t)l * NRBF;
    const float* rb    = rbf_betas + (size_t)l * NRBF;
    const float* w1    = eo_w1 + (size_t)l * HEIN * HDIM;
    const float* b1    = eo_b1 + (size_t)l * HDIM;
    const float* w2    = eo_w2 + (size_t)l * HDIM * HDIM;
    const float* b2    = eo_b2 + (size_t)l * HDIM;
    const float* sw    = sem_w + (size_t)l * HDIM * NHEADS;
    const float* sb    = sem_b + (size_t)l * NHEADS;
    const float* xw    = xmix_w + (size_t)l * NCOEF * NCOEF;
    const float* p1    = pn_w1 + (size_t)l * NCOEF * HDIM;
    const float* pb1   = pn_b1 + (size_t)l * HDIM;
    const float* p2    = pn_w2 + (size_t)l * HDIM * HDIM;
    const float* pb2   = pn_b2 + (size_t)l * HDIM;
    const float* nw1   = node_w1 + (size_t)l * NODEIN * HDIM;
    const float* nb1   = node_b1 + (size_t)l * HDIM;
    const float* nw2   = node_w2 + (size_t)l * HDIM * HDIM;
    const float* nb2   = node_b2 + (size_t)l * HDIM;

    // hc gather -> he_in[:,0:128]
    k_gather<<<blk(NE * 128), 256, 0, stream>>>(edges, hbuf, hein);
    // hm = hc @ edge_in_w + b   (K=128 reads only cols 0..127 of he_in)
    gemm(hein, HEINP, ei_w, NRBF, ei_b, nullptr, hm, 64, NE, NRBF, 128, 0);
    // he_in[:,128:192] = rbf*hm | d | zeros
    k_rbf<<<blk(NE * 64), 256, 0, stream>>>(hm, rbfb, dbuf, rm, rb, hein);
    // t1 = silu(he_in @ W1 + b1)   (t1 overwrites hm)
    gemm(hein, HEINP, w1, HDIM, b1, nullptr, hm, HDIM, NE, HDIM, HEIN, 1);
    // he = t1 @ W2 + b2
    gemm(hm, HDIM, w2, HDIM, b2, nullptr, he, HDIM, NE, HDIM, HDIM, 0);
    // attention logits (celu)
    k_sem<<<blk(NE * NHEADS), 256, 0, stream>>>(he, sw, sb, logits);
    // zero attention accumulators + node accumulators
    k_fill0<<<blk(att_cnt), 256, 0, stream>>>((float*)nmax_s, att_cnt);
    k_fill0<<<blk(node_cnt), 256, 0, stream>>>(csum, node_cnt);
    // double segment softmax + joint renormalization
    k_att_max<<<blk(NE * NHEADS), 256, 0, stream>>>(edges, logits, dbuf, nmax_s, nmax_e);
    k_att_exp<<<blk(NE * NHEADS), 256, 0, stream>>>(edges, logits, dbuf, nmax_s, nmax_e,
                                                    es, ee, nsum_s, nsum_e);
    k_att_un<<<blk(NE * NHEADS), 256, 0, stream>>>(edges, es, ee, nsum_s, nsum_e, att, natt);
    // hea = outer(he, att_norm)
    k_hea<<<blk(NE * NCOEF), 256, 0, stream>>>(edges, he, att, natt, hea);
    // coef = tanh(hea @ xmix)
    gemm(hea, NCOEF, xw, NCOEF, nullptr, nullptr, coefb, NCOEF, NE, NCOEF, NCOEF, 2);
    // segment sums: h_agg, csum
    k_scatter<<<blk(NE * NCOEF), 256, 0, stream>>>(edges, hea, coefb, dirs, hagg, csum);
    // cnorm = |csum|^2
    k_cnorm<<<blk(NN * NCOEF), 256, 0, stream>>>(csum, cnorm);
    // h_comb = silu(silu(cnorm @ pn_w1 + b) @ pn_w2 + b)
    gemm(cnorm, NCOEF, p1, HDIM, pb1, nullptr, tnode, HDIM, NN, HDIM, NCOEF, 1);
    gemm(tnode, HDIM, p2, HDIM, pb2, nullptr, hcomb, HDIM, NN, HDIM, HDIM, 1);
    // upd_in = [h | h_agg | h_comb]
    k_updin<<<blk(NN * NODEIN), 256, 0, stream>>>(hbuf, hagg, hcomb, updin);
    // h = h + silu(silu(upd_in @ nw1 + b) @ nw2 + b)
    gemm(updin, NODEIN, nw1, HDIM, nb1, nullptr, tnode, HDIM, NN, HDIM, NODEIN, 1);
    gemm(tnode, HDIM, nw2, HDIM, nb2, hbuf, hbuf, HDIM, NN, HDIM, HDIM, 1);
  }

  // final embedding MLP
  gemm(hbuf, HDIM, emo_w1, HDIM, emo_b1, nullptr, tnode, HDIM, NN, HDIM, HDIM, 1);
  gemm(tnode, HDIM, emo_w2, HDIM, emo_b2, nullptr, hfin, HDIM, NN, HDIM, HDIM, 0);
  // graph pooling
  k_fill0<<<blk(NSEGS * HDIM), 256, 0, stream>>>(ybuf, NSEGS * HDIM);
  k_yscatter<<<blk(NN * HDIM), 256, 0, stream>>>(segs, hfin, ybuf);
  // readout head
  k_head<<<dim3(1), 128, 0, stream>>>(ybuf, head_w1, head_b1, head_w2, head_b2, (float*)d_out);

  (void)in_sizes; (void)n_in; (void)out_size; (void)ws_size;
}